// VariationalGCNEncoder_86474871537723
// MI455X (gfx1250) — compile-verified
//
#include <hip/hip_runtime.h>
#include <hip/hip_bf16.h>

typedef float v2f __attribute__((ext_vector_type(2)));
typedef float v8f __attribute__((ext_vector_type(8)));

#define WPB 8  // waves per block (256 threads, wave32)

// ---------- small prep kernels ----------

__global__ __launch_bounds__(256) void init_one_kernel(float* p, int n) {
  int i = blockIdx.x * 256 + threadIdx.x;
  if (i < n) p[i] = 1.0f;  // self-loop contributes 1 to every degree
}

__global__ __launch_bounds__(256) void degree_kernel(const long long* __restrict__ ei,
                                                     float* __restrict__ deg, int E) {
  int e = blockIdx.x * 256 + threadIdx.x;
  if (e < E) atomicAdd(deg + (int)ei[(size_t)E + e], 1.0f);
}

__global__ __launch_bounds__(256) void rsqrt_kernel(float* p, int n) {
  int i = blockIdx.x * 256 + threadIdx.x;
  if (i < n) p[i] = rsqrtf(p[i]);  // deg >= 1 always (self loops)
}

// Wcat[k][c] = c<64 ? W_mu[k][c] : W_ls[k][c-64]   (128x128 row-major)
__global__ __launch_bounds__(256) void build_wcat_kernel(const float* __restrict__ Wmu,
                                                         const float* __restrict__ Wls,
                                                         float* __restrict__ Wcat) {
  int i = blockIdx.x * 256 + threadIdx.x;  // 0..16383
  int k = i >> 7, c = i & 127;
  Wcat[i] = (c < 64) ? Wmu[k * 64 + c] : Wls[k * 64 + (c - 64)];
}

// ---------- fp32 WMMA GEMM: C[nrows,128] = A[nrows,128] x B[128,128] ----------
// One wave computes a 16-row x 128-col strip: 8 accumulator tiles, K-loop of
// 32 steps of V_WMMA_F32_16X16X4_F32.
__global__ __launch_bounds__(256) void gemm128_wmma_kernel(const float* __restrict__ A,
                                                           const float* __restrict__ B,
                                                           float* __restrict__ C,
                                                           int nrows) {
  __shared__ float Bs[128 * 128];  // 64 KB, row-major copy of B
  for (int i = threadIdx.x; i < 4096; i += 256)
    reinterpret_cast<float4*>(Bs)[i] = reinterpret_cast<const float4*>(B)[i];
  __syncthreads();

  const int wave = threadIdx.x >> 5;
  const int lane = threadIdx.x & 31;
  const int m0 = (blockIdx.x * WPB + wave) * 16;
  if (m0 >= nrows) return;  // wave-uniform: EXEC stays all-ones for WMMA

  const int mlane = lane & 15;        // A row within tile / C col within tile
  const int khalf = (lane >> 4) << 1; // 0 or 2: K-pair selector (ISA 16x4 f32 A layout)

  int arow = m0 + mlane;
  if (arow >= nrows) arow = nrows - 1;  // clamp (N is a multiple of 16 anyway)
  const float* Ap = A + (size_t)arow * 128 + khalf;

  v8f acc[8] = {};

  for (int kk = 0; kk < 128; kk += 4) {
    v2f av = *reinterpret_cast<const v2f*>(Ap + kk);  // A[m][kk+khalf .. +1]
    const int kb = kk + khalf;
#pragma unroll
    for (int t = 0; t < 8; t++) {
      const int c = t * 16 + mlane;
      v2f bv;
      bv.x = Bs[kb * 128 + c];        // B[kb][c]
      bv.y = Bs[kb * 128 + 128 + c];  // B[kb+1][c]
      acc[t] = __builtin_amdgcn_wmma_f32_16x16x4_f32(
          /*neg_a=*/false, av, /*neg_b=*/false, bv,
          /*c_mod=*/(short)0, acc[t], /*reuse_a=*/false, /*reuse_b=*/false);
    }
  }

  // C/D layout: VGPR r, lanes 0-15 -> M=r, lanes 16-31 -> M=r+8; N = lane&15
  const int rbase = m0 + ((lane >> 4) << 3);
#pragma unroll
  for (int t = 0; t < 8; t++) {
#pragma unroll
    for (int r = 0; r < 8; r++) {
      int row = rbase + r;
      if (row < nrows) C[(size_t)row * 128 + t * 16 + mlane] = acc[t][r];
    }
  }
}

// ---------- edge aggregation (wave per edge, float4 per lane) ----------
// items [0,E): real edges; items [E,E+N): self loops (s=d=item-E).

__global__ __launch_bounds__(256) void aggregate_h_kernel(const long long* __restrict__ ei,
                                                          const float* __restrict__ dinv,
                                                          const float* __restrict__ feat,
                                                          float* __restrict__ out,
                                                          int E, int N) {
  int item = blockIdx.x * WPB + (threadIdx.x >> 5);
  int lane = threadIdx.x & 31;
  if (item >= E + N) return;
  int s, d;
  if (item < E) { s = (int)ei[item]; d = (int)ei[(size_t)E + item]; }
  else          { s = d = item - E; }
  float w = dinv[s] * dinv[d];
  float4 v = reinterpret_cast<const float4*>(feat + (size_t)s * 128)[lane];
  float* op = out + (size_t)d * 128 + lane * 4;
  atomicAdd(op + 0, w * v.x);
  atomicAdd(op + 1, w * v.y);
  atomicAdd(op + 2, w * v.z);
  atomicAdd(op + 3, w * v.w);
}

// second stage: feat=[N,128] = [h@W_mu | h@W_ls]; scatter into split d_out
__global__ __launch_bounds__(256) void aggregate_out_kernel(const long long* __restrict__ ei,
                                                            const float* __restrict__ dinv,
                                                            const float* __restrict__ feat,
                                                            float* __restrict__ out,
                                                            int E, int N) {
  int item = blockIdx.x * WPB + (threadIdx.x >> 5);
  int lane = threadIdx.x & 31;
  if (item >= E + N) return;
  int s, d;
  if (item < E) { s = (int)ei[item]; d = (int)ei[(size_t)E + item]; }
  else          { s = d = item - E; }
  float w = dinv[s] * dinv[d];
  float4 v = reinterpret_cast<const float4*>(feat + (size_t)s * 128)[lane];
  // lanes 0-15 -> mu (cols 0..63), lanes 16-31 -> logstd (cols 64..127)
  float* op = (lane < 16)
      ? out + (size_t)d * 64 + lane * 4
      : out + (size_t)N * 64 + (size_t)d * 64 + (lane - 16) * 4;
  atomicAdd(op + 0, w * v.x);
  atomicAdd(op + 1, w * v.y);
  atomicAdd(op + 2, w * v.z);
  atomicAdd(op + 3, w * v.w);
}

// ---------- bias / activation ----------

__global__ __launch_bounds__(256) void bias_relu_kernel(float* __restrict__ h,
                                                        const float* __restrict__ b,
                                                        size_t total) {
  size_t i = (size_t)blockIdx.x * 256 + threadIdx.x;
  if (i < total) {
    float v = h[i] + b[i & 127];
    h[i] = v > 0.0f ? v : 0.0f;
  }
}

__global__ __launch_bounds__(256) void out_bias_kernel(float* __restrict__ out,
                                                       const float* __restrict__ bmu,
                                                       const float* __restrict__ bls,
                                                       size_t half) {
  size_t i = (size_t)blockIdx.x * 256 + threadIdx.x;
  if (i < half) {
    out[i]        += bmu[i & 63];
    out[half + i] += bls[i & 63];
  }
}

// ---------- launch ----------

extern "C" void kernel_launch(void* const* d_in, const int* in_sizes, int n_in,
                              void* d_out, int out_size, void* d_ws, size_t ws_size,
                              hipStream_t stream) {
  const float* x   = (const float*)d_in[0];
  const float* W1  = (const float*)d_in[1];
  const float* b1  = (const float*)d_in[2];
  const float* Wmu = (const float*)d_in[3];
  const float* bmu = (const float*)d_in[4];
  const float* Wls = (const float*)d_in[5];
  const float* bls = (const float*)d_in[6];
  const long long* ei = (const long long*)d_in[7];
  float* out = (float*)d_out;

  const int N = in_sizes[0] / 128;  // 100000
  const int E = in_sizes[7] / 2;    // 1600000

  // workspace layout (floats): dinv | xw/hw2 | h | Wcat  (~103 MB)
  float* ws   = (float*)d_ws;
  float* dinv = ws;                                 // N  (padded slot of 102400)
  float* bufA = ws + 102400;                        // N*128 : xw, then reused for [h@Wmu|h@Wls]
  float* bufH = bufA + (size_t)N * 128;             // N*128 : aggregated h
  float* Wcat = bufH + (size_t)N * 128;             // 128*128

  // zero accumulation targets every call (graph-replay safe)
  hipMemsetAsync(bufH, 0, (size_t)N * 128 * sizeof(float), stream);
  hipMemsetAsync(out, 0, (size_t)out_size * sizeof(float), stream);

  init_one_kernel<<<(N + 255) / 256, 256, 0, stream>>>(dinv, N);
  degree_kernel<<<(E + 255) / 256, 256, 0, stream>>>(ei, dinv, E);
  rsqrt_kernel<<<(N + 255) / 256, 256, 0, stream>>>(dinv, N);
  build_wcat_kernel<<<64, 256, 0, stream>>>(Wmu, Wls, Wcat);

  const int strips  = (N + 15) / 16;
  const int gblocks = (strips + WPB - 1) / WPB;
  gemm128_wmma_kernel<<<gblocks, 256, 0, stream>>>(x, W1, bufA, N);

  const int items   = E + N;
  const int ablocks = (items + WPB - 1) / WPB;
  aggregate_h_kernel<<<ablocks, 256, 0, stream>>>(ei, dinv, bufA, bufH, E, N);

  const size_t tot = (size_t)N * 128;
  bias_relu_kernel<<<(int)((tot + 255) / 256), 256, 0, stream>>>(bufH, b1, tot);

  gemm128_wmma_kernel<<<gblocks, 256, 0, stream>>>(bufH, Wcat, bufA, N);
  aggregate_out_kernel<<<ablocks, 256, 0, stream>>>(ei, dinv, bufA, out, E, N);

  const size_t half = (size_t)N * 64;
  out_bias_kernel<<<(int)((half + 255) / 256), 256, 0, stream>>>(out, bmu, bls, half);
}